// AttentionDecoder_70626442216037
// MI455X (gfx1250) — compile-verified
//
#include <hip/hip_runtime.h>
#include <hip/hip_bf16.h>
#include <math.h>

#define OUT_SZ    32000
#define EMBED_SZ  2048
#define HIDDEN_SZ 2048
#define N_LAYERS  2
#define ATTN_LEN  50

typedef float v2f __attribute__((ext_vector_type(2)));
typedef float v8f __attribute__((ext_vector_type(8)));

// D(16x16,f32) = A(16x4,f32) x B(4x16,f32) + C ; wave32, EXEC all-ones.
#define WMMA4(a, b, c) \
  __builtin_amdgcn_wmma_f32_16x16x4_f32(false, (a), false, (b), (short)0, (c), false, false)

__device__ __forceinline__ float sig_(float x) { return 1.0f / (1.0f + expf(-x)); }

// ---------------------------------------------------------------------------
// Attention scores: one block per encoder position j (50 blocks).
// score[j] = attn_W[j,:2048] @ emb + attn_W[j,2048:] @ h0 + attn_b[j]
// ---------------------------------------------------------------------------
__global__ void __launch_bounds__(256) attn_score_kernel(
    const int* __restrict__ in_idx,
    const float* __restrict__ embed_table,
    const float* __restrict__ hidden0,
    const float* __restrict__ attn_W,
    const float* __restrict__ attn_b,
    float* __restrict__ scores)
{
  __shared__ float red[256];
  const int tid = threadIdx.x;
  const int j   = blockIdx.x;
  const float* emb = embed_table + (size_t)in_idx[0] * EMBED_SZ;
  const float* wr  = attn_W + (size_t)j * (EMBED_SZ + HIDDEN_SZ);

  float acc = 0.0f;
  for (int k = tid; k < EMBED_SZ; k += 256)  acc += wr[k] * emb[k];
  for (int k = tid; k < HIDDEN_SZ; k += 256) acc += wr[EMBED_SZ + k] * hidden0[k];
  red[tid] = acc;
  __syncthreads();
  for (int s = 128; s > 0; s >>= 1) {
    if (tid < s) red[tid] += red[tid + s];
    __syncthreads();
  }
  if (tid == 0) scores[j] = red[0] + attn_b[j];
}

// ---------------------------------------------------------------------------
// Softmax over the 50 scores (tiny, serial in one lane -> deterministic).
// Writes weights both to workspace (for context kernel) and d_out tail.
// ---------------------------------------------------------------------------
__global__ void __launch_bounds__(64) attn_softmax_kernel(
    const float* __restrict__ scores,
    float* __restrict__ w_ws,
    float* __restrict__ w_out)
{
  __shared__ float sc[ATTN_LEN];
  const int tid = threadIdx.x;
  if (tid < ATTN_LEN) sc[tid] = scores[tid];
  __syncthreads();
  if (tid == 0) {
    float m = sc[0];
    for (int j = 1; j < ATTN_LEN; ++j) m = fmaxf(m, sc[j]);
    float s = 0.0f;
    for (int j = 0; j < ATTN_LEN; ++j) { float e = expf(sc[j] - m); sc[j] = e; s += e; }
    float inv = 1.0f / s;
    for (int j = 0; j < ATTN_LEN; ++j) sc[j] *= inv;
  }
  __syncthreads();
  if (tid < ATTN_LEN) { w_ws[tid] = sc[tid]; w_out[tid] = sc[tid]; }
}

// ---------------------------------------------------------------------------
// Context vector + staging of emb into xin:
// xin[k] = emb[k]; xin[2048+k] = sum_j w[j]*enc[j,k]. 8 blocks x 256 threads.
// ---------------------------------------------------------------------------
__global__ void __launch_bounds__(256) attn_context_kernel(
    const int* __restrict__ in_idx,
    const float* __restrict__ embed_table,
    const float* __restrict__ enc,
    const float* __restrict__ w,
    float* __restrict__ xin)
{
  const int k = blockIdx.x * 256 + threadIdx.x;
  const float* emb = embed_table + (size_t)in_idx[0] * EMBED_SZ;
  float acc = 0.0f;
#pragma unroll 10
  for (int j = 0; j < ATTN_LEN; ++j) acc += w[j] * enc[(size_t)j * HIDDEN_SZ + k];
  xin[k] = emb[k];
  xin[EMBED_SZ + k] = acc;
}

// ---------------------------------------------------------------------------
// WMMA fp32 GEMV: 16 rows per block-column, optional K-split via gridDim.y.
// Block = 8 waves; wave w covers a K/(8*kdiv) slice with 16x16x4 f32 WMMA.
// A lane map (ISA 7.12.2): lane&15 = M, (lane>>4)*2 = K offset {0,2}; the v2f
// holds K,K+1 of that half. B = x broadcast over N with the same K<->slot map,
// so the K pairing is layout-consistent. Column N=0 of D (lanes 0/16, VGPRs
// 0..7) is the GEMV result; deterministic LDS cross-wave reduction.
// ---------------------------------------------------------------------------
__global__ void __launch_bounds__(256) wmma_gemv_kernel(
    const float* __restrict__ W, const float* __restrict__ x,
    const float* __restrict__ bias, float* __restrict__ y,
    float* __restrict__ part, int K)
{
  const int tid    = threadIdx.x;
  const int lane   = tid & 31;
  const int wave_u = __builtin_amdgcn_readfirstlane(tid >> 5);  // wave-uniform
  const int row0   = blockIdx.x << 4;
  const int nrows  = gridDim.x << 4;
  const int r      = lane & 15;
  const int koff   = (lane >> 4) << 1;          // 0 or 2
  const int kdiv   = gridDim.y;
  const int kblock = K / kdiv;                  // per block
  const int kslice = kblock >> 3;               // per wave (multiple of 32)
  const int k0     = blockIdx.y * kblock + wave_u * kslice;

  const float* wrow = W + (size_t)(row0 + r) * K + koff;
  const float* xv   = x + koff;

  v8f acc0 = {}, acc1 = {};
#pragma unroll 4
  for (int k = k0; k < k0 + kslice; k += 8) {
    v2f a0 = __builtin_nontemporal_load((const v2f*)(wrow + k));
    v2f b0 = *(const v2f*)(xv + k);
    acc0 = WMMA4(a0, b0, acc0);
    v2f a1 = __builtin_nontemporal_load((const v2f*)(wrow + k + 4));
    v2f b1 = *(const v2f*)(xv + k + 4);
    acc1 = WMMA4(a1, b1, acc1);
  }

  __shared__ float partial[128];
  if (lane == 0 || lane == 16) {
    const int base = wave_u * 16 + ((lane >> 4) << 3);  // rows 0..7 / 8..15
#pragma unroll
    for (int g = 0; g < 8; ++g) partial[base + g] = acc0[g] + acc1[g];
  }
  __syncthreads();
  if (tid < 16) {
    float s = 0.0f;
#pragma unroll
    for (int w = 0; w < 8; ++w) s += partial[w * 16 + tid];
    const int row = row0 + tid;
    if (kdiv == 1) y[row] = s + bias[row];
    else           part[blockIdx.y * nrows + row] = s;
  }
}

// Combine K-split partials (deterministic fixed order) and add bias.
__global__ void __launch_bounds__(256) gemv_combine_kernel(
    const float* __restrict__ part, const float* __restrict__ bias,
    float* __restrict__ y, int nrows, int kdiv)
{
  const int r = blockIdx.x * 256 + threadIdx.x;
  float s = bias[r];
  for (int i = 0; i < kdiv; ++i) s += part[i * nrows + r];
  y[r] = s;
}

// ---------------------------------------------------------------------------
// LSTM gates: gates[r] = Wih[r,:]@x + Whh[r,:]@h + bih[r] + bhh[r].
// 16 rows/block; waves 0-3 cover Wih (512-wide K slices), waves 4-7 Whh.
// ---------------------------------------------------------------------------
__global__ void __launch_bounds__(256) lstm_gates_kernel(
    const float* __restrict__ Wih, const float* __restrict__ Whh,
    const float* __restrict__ x, const float* __restrict__ h,
    const float* __restrict__ bih, const float* __restrict__ bhh,
    float* __restrict__ gates)
{
  const int tid    = threadIdx.x;
  const int lane   = tid & 31;
  const int wave_u = __builtin_amdgcn_readfirstlane(tid >> 5);
  const int row0   = blockIdx.x << 4;
  const int r      = lane & 15;
  const int koff   = (lane >> 4) << 1;

  const float* W = (wave_u < 4) ? Wih : Whh;
  const float* v = (wave_u < 4) ? x : h;
  const int k0 = (wave_u & 3) * (HIDDEN_SZ / 4);

  const float* wrow = W + (size_t)(row0 + r) * HIDDEN_SZ + koff;
  const float* xv   = v + koff;

  v8f acc0 = {}, acc1 = {};
#pragma unroll 4
  for (int k = k0; k < k0 + (HIDDEN_SZ / 4); k += 8) {
    v2f a0 = __builtin_nontemporal_load((const v2f*)(wrow + k));
    v2f b0 = *(const v2f*)(xv + k);
    acc0 = WMMA4(a0, b0, acc0);
    v2f a1 = __builtin_nontemporal_load((const v2f*)(wrow + k + 4));
    v2f b1 = *(const v2f*)(xv + k + 4);
    acc1 = WMMA4(a1, b1, acc1);
  }

  __shared__ float partial[128];
  if (lane == 0 || lane == 16) {
    const int base = wave_u * 16 + ((lane >> 4) << 3);
#pragma unroll
    for (int g = 0; g < 8; ++g) partial[base + g] = acc0[g] + acc1[g];
  }
  __syncthreads();
  if (tid < 16) {
    float s = bih[row0 + tid] + bhh[row0 + tid];
#pragma unroll
    for (int w = 0; w < 8; ++w) s += partial[w * 16 + tid];
    gates[row0 + tid] = s;
  }
}

// ---------------------------------------------------------------------------
// LSTM pointwise (PyTorch gate order i,f,g,o).
// ---------------------------------------------------------------------------
__global__ void __launch_bounds__(256) lstm_pointwise_kernel(
    const float* __restrict__ gates, const float* __restrict__ c_in,
    float* __restrict__ h_out, float* __restrict__ c_out,
    float* __restrict__ xnext)
{
  const int k = blockIdx.x * 256 + threadIdx.x;
  const float ig = sig_(gates[k]);
  const float fg = sig_(gates[HIDDEN_SZ + k]);
  const float gg = tanhf(gates[2 * HIDDEN_SZ + k]);
  const float og = sig_(gates[3 * HIDDEN_SZ + k]);
  const float c  = fg * c_in[k] + ig * gg;
  const float hh = og * tanhf(c);
  c_out[k] = c;
  h_out[k] = hh;
  xnext[k] = hh;
}

// ---------------------------------------------------------------------------
// log-softmax over 32000: parallel online logsumexp.
// Combine rule for (m,s) pairs: M=max(m1,m2); S=s1*e^(m1-M)+s2*e^(m2-M).
// ---------------------------------------------------------------------------
__global__ void __launch_bounds__(256) lse_partial_kernel(
    const float* __restrict__ logits, float* __restrict__ pm, float* __restrict__ ps)
{
  __shared__ float rm[256], rs[256];
  const int tid = threadIdx.x;
  const int i   = blockIdx.x * 256 + tid;
  rm[tid] = logits[i];
  rs[tid] = 1.0f;
  __syncthreads();
  for (int s = 128; s > 0; s >>= 1) {
    if (tid < s) {
      const float m1 = rm[tid], s1 = rs[tid];
      const float m2 = rm[tid + s], s2 = rs[tid + s];
      const float M = fmaxf(m1, m2);
      rm[tid] = M;
      rs[tid] = s1 * expf(m1 - M) + s2 * expf(m2 - M);
    }
    __syncthreads();
  }
  if (tid == 0) { pm[blockIdx.x] = rm[0]; ps[blockIdx.x] = rs[0]; }
}

__global__ void __launch_bounds__(128) lse_final_kernel(
    const float* __restrict__ pm, const float* __restrict__ ps,
    int n, float* __restrict__ scal)
{
  __shared__ float rm[128], rs[128];
  const int tid = threadIdx.x;
  if (tid < n) { rm[tid] = pm[tid]; rs[tid] = ps[tid]; }
  else         { rm[tid] = -3.0e38f; rs[tid] = 0.0f; }
  __syncthreads();
  for (int s = 64; s > 0; s >>= 1) {
    if (tid < s) {
      const float m1 = rm[tid], s1 = rs[tid];
      const float m2 = rm[tid + s], s2 = rs[tid + s];
      const float M = fmaxf(m1, m2);
      rm[tid] = M;
      rs[tid] = s1 * expf(m1 - M) + s2 * expf(m2 - M);
    }
    __syncthreads();
  }
  if (tid == 0) scal[0] = rm[0] + logf(rs[0]);
}

__global__ void __launch_bounds__(256) lsm_sub_kernel(
    const float* __restrict__ logits, const float* __restrict__ scal,
    float* __restrict__ out)
{
  const int i = blockIdx.x * 256 + threadIdx.x;
  out[i] = logits[i] - scal[0];
}

// ---------------------------------------------------------------------------
extern "C" void kernel_launch(void* const* d_in, const int* in_sizes, int n_in,
                              void* d_out, int out_size, void* d_ws, size_t ws_size,
                              hipStream_t stream) {
  const int*   in_idx = (const int*)d_in[0];
  const float* hidden = (const float*)d_in[1];   // [2,1,2048]
  const float* cell   = (const float*)d_in[2];   // [2,1,2048]
  const float* enc    = (const float*)d_in[3];   // [50,2048]
  const float* embed  = (const float*)d_in[4];   // [32000,2048]
  const float* attn_W = (const float*)d_in[5];   // [50,4096]
  const float* attn_b = (const float*)d_in[6];   // [50]
  const float* comb_W = (const float*)d_in[7];   // [2048,4096]
  const float* comb_b = (const float*)d_in[8];   // [2048]
  const float* Wih    = (const float*)d_in[9];   // [2,8192,2048]
  const float* Whh    = (const float*)d_in[10];  // [2,8192,2048]
  const float* bih    = (const float*)d_in[11];  // [2,8192]
  const float* bhh    = (const float*)d_in[12];  // [2,8192]
  const float* out_W  = (const float*)d_in[13];  // [32000,2048]
  const float* out_b  = (const float*)d_in[14];  // [32000]

  // d_out: [log_softmax 32000][h 2*2048][c 2*2048][attn_weights 50]
  float* out    = (float*)d_out;
  float* out_h  = out + OUT_SZ;
  float* out_c  = out + OUT_SZ + N_LAYERS * HIDDEN_SZ;
  float* out_aw = out + OUT_SZ + 2 * N_LAYERS * HIDDEN_SZ;

  // workspace layout (floats)
  float* ws      = (float*)d_ws;
  float* xin     = ws;             // 4096
  float* xbuf    = ws + 4096;      // 2048
  float* gates   = ws + 6144;      // 8192
  float* logits  = ws + 14336;     // 32000
  float* scores  = ws + 46336;     // 64
  float* attnw   = ws + 46400;     // 64
  float* cpart   = ws + 46464;     // 2*2048 comb K-split partials
  float* lse_m   = ws + 50560;     // 128
  float* lse_s   = ws + 50688;     // 128
  float* scal    = ws + 50816;     // 1

  // ---- attention ----
  attn_score_kernel<<<ATTN_LEN, 256, 0, stream>>>(in_idx, embed, hidden,
                                                  attn_W, attn_b, scores);
  attn_softmax_kernel<<<1, 64, 0, stream>>>(scores, attnw, out_aw);
  attn_context_kernel<<<HIDDEN_SZ / 256, 256, 0, stream>>>(in_idx, embed, enc,
                                                           attnw, xin);

  // ---- x = comb_W @ [emb|context] + comb_b  (K=4096, K-split 2) ----
  wmma_gemv_kernel<<<dim3(HIDDEN_SZ / 16, 2), 256, 0, stream>>>(
      comb_W, xin, comb_b, xbuf, cpart, EMBED_SZ + HIDDEN_SZ);
  gemv_combine_kernel<<<HIDDEN_SZ / 256, 256, 0, stream>>>(cpart, comb_b, xbuf,
                                                           HIDDEN_SZ, 2);

  // ---- 2-layer LSTM step ----
  for (int l = 0; l < N_LAYERS; ++l) {
    const size_t wstride = (size_t)4 * HIDDEN_SZ * HIDDEN_SZ;
    lstm_gates_kernel<<<(4 * HIDDEN_SZ) / 16, 256, 0, stream>>>(
        Wih + (size_t)l * wstride, Whh + (size_t)l * wstride,
        xbuf, hidden + l * HIDDEN_SZ,
        bih + l * 4 * HIDDEN_SZ, bhh + l * 4 * HIDDEN_SZ, gates);
    lstm_pointwise_kernel<<<HIDDEN_SZ / 256, 256, 0, stream>>>(
        gates, cell + l * HIDDEN_SZ,
        out_h + l * HIDDEN_SZ, out_c + l * HIDDEN_SZ, xbuf);
  }

  // ---- logits = out_W @ h + out_b (K=2048, 2000 blocks) ----
  wmma_gemv_kernel<<<dim3(OUT_SZ / 16, 1), 256, 0, stream>>>(
      out_W, xbuf, out_b, logits, cpart, HIDDEN_SZ);

  // ---- log_softmax, parallel ----
  lse_partial_kernel<<<OUT_SZ / 256, 256, 0, stream>>>(logits, lse_m, lse_s);
  lse_final_kernel<<<1, 128, 0, stream>>>(lse_m, lse_s, OUT_SZ / 256, scal);
  lsm_sub_kernel<<<OUT_SZ / 256, 256, 0, stream>>>(logits, scal, out);
}